// SimpleGAT_30537217475375
// MI455X (gfx1250) — compile-verified
//
#include <hip/hip_runtime.h>
#include <hip/hip_bf16.h>
#include <math.h>

// ---------------------------------------------------------------------------
// Problem constants (match reference)
// ---------------------------------------------------------------------------
#define NN      50000      // nodes
#define NE      800000     // raw edges
#define ET      850000     // edges incl. self loops
#define F_IN    128
#define HIDF    64
#define NCLS    10
#define NEG_SLOPE 0.2f

typedef float v2f __attribute__((ext_vector_type(2)));
typedef float v8f __attribute__((ext_vector_type(8)));

// ---------------------------------------------------------------------------
// Workspace layout (float elements)
// ---------------------------------------------------------------------------
#define OFF_H     ((size_t)0)                    // 50000*64 : transformed feats / lin0 out
#define OFF_AGGA  ((size_t)NN * 64)              // 50000*64 : GAT layer-0 aggregate
#define OFF_AGGB  ((size_t)2 * NN * 64)          // 50000*64 : GAT layer-1 aggregate
#define OFF_SS    ((size_t)3 * NN * 64)          // 50000    : s_src
#define OFF_SD    (OFF_SS + NN)                  // 50000    : s_dst
#define OFF_M     (OFF_SD + NN)                  // 50000    : segment max (ordered uint)
#define OFF_DEN   (OFF_M + NN)                   // 50000    : softmax denom
#define OFF_EX    (OFF_DEN + NN)                 // 850000   : exp(e - max) per edge

#define ORD_NEG_INF 0x007FFFFFu                  // f2ord(-inf)

// monotonic float <-> uint mapping (order-preserving)
__device__ __forceinline__ unsigned f2ord(float f) {
    unsigned u = __float_as_uint(f);
    return (u & 0x80000000u) ? ~u : (u | 0x80000000u);
}
__device__ __forceinline__ float ord2f(unsigned u) {
    return (u & 0x80000000u) ? __uint_as_float(u & 0x7FFFFFFFu)
                             : __uint_as_float(~u);
}

__device__ __forceinline__ void edge_nodes(int e, const int* __restrict__ ei,
                                           int& s, int& d) {
    if (e < NE) { s = ei[e]; d = ei[NE + e]; }
    else        { s = e - NE; d = e - NE; }          // self loop
}

// ---------------------------------------------------------------------------
// WMMA fp32 GEMM:  C[M x NCOLS] = (A[M x K] + abias) * B[K x NCOLS]
// N padded to 16*NT. B staged transposed + zero-padded in LDS so every
// fragment load is one aligned ds_load_b64 and the k-loop is branch-free.
// One wave computes a 16-row strip across all NT 16-col tiles.
// ---------------------------------------------------------------------------
template <int NT, int K, int NCOLS, bool ABIAS, bool BIAS_RELU>
__global__ __launch_bounds__(256) void gemm_wmma_kernel(
    const float* __restrict__ A, const float* __restrict__ B,
    const float* __restrict__ abias, const float* __restrict__ cbias,
    float* __restrict__ C, int M)
{
    __shared__ float Bt[NT * 16 * K];            // [col][k], col-major-transposed
    const int tid = threadIdx.x;
    for (int idx = tid; idx < NT * 16 * K; idx += 256) {
        const int col = idx / K;
        const int k   = idx - col * K;
        Bt[col * K + k] = (col < NCOLS) ? B[k * NCOLS + col] : 0.f;
    }
    __syncthreads();

    const int wave = blockIdx.x * 8 + (tid >> 5);
    const int lane = tid & 31;
    const int strips = M >> 4;                   // M is a multiple of 16
    if (wave >= strips) return;

    const int row0 = wave << 4;
    const int r  = lane & 15;                    // A row / B,C col within tile
    const int kb = (lane >> 4) << 1;             // K sub-offset: 0 or 2
    const float* __restrict__ Arow = A + (size_t)(row0 + r) * K + kb;

    v8f acc[NT] = {};
#pragma unroll
    for (int k0 = 0; k0 < K; k0 += 4) {
        v2f a = *(const v2f*)(Arow + k0);        // global_load_b64 (aligned)
        if (ABIAS) {
            a.x += abias[k0 + kb];
            a.y += abias[k0 + kb + 1];
        }
#pragma unroll
        for (int t = 0; t < NT; ++t) {
            v2f b = *(const v2f*)(&Bt[(t * 16 + r) * K + k0 + kb]); // ds_load_b64
            acc[t] = __builtin_amdgcn_wmma_f32_16x16x4_f32(
                false, a, false, b, (short)0, acc[t], false, false);
        }
    }

    const int col  = lane & 15;
    const int rAdd = (lane < 16) ? 0 : 8;
#pragma unroll
    for (int rr = 0; rr < 8; ++rr) {
        const int row = row0 + rr + rAdd;
#pragma unroll
        for (int t = 0; t < NT; ++t) {
            const int c = t * 16 + col;
            if (c < NCOLS) {                     // folds away when 16*NT == NCOLS
                float v = acc[t][rr];
                if (BIAS_RELU) { v += cbias[c]; v = fmaxf(v, 0.f); }
                C[(size_t)row * NCOLS + c] = v;
            }
        }
    }
}

// ---------------------------------------------------------------------------
// Per-node attention partial scores. One wave per node; 2 feats/lane via b64.
// ---------------------------------------------------------------------------
__global__ __launch_bounds__(256) void scores_kernel(
    const float* __restrict__ h, const float* __restrict__ att_src,
    const float* __restrict__ att_dst, float* __restrict__ s_src,
    float* __restrict__ s_dst, int n)
{
    const int wave = (blockIdx.x * blockDim.x + threadIdx.x) >> 5;
    const int lane = threadIdx.x & 31;
    if (wave >= n) return;
    v2f as = *(const v2f*)(att_src + 2 * lane);
    v2f ad = *(const v2f*)(att_dst + 2 * lane);
    v2f hv = *(const v2f*)(h + (size_t)wave * 64 + 2 * lane);
    float ds = hv.x * as.x + hv.y * as.y;
    float dd = hv.x * ad.x + hv.y * ad.y;
#pragma unroll
    for (int m = 16; m >= 1; m >>= 1) {
        ds += __shfl_xor(ds, m, 32);
        dd += __shfl_xor(dd, m, 32);
    }
    if (lane == 0) { s_src[wave] = ds; s_dst[wave] = dd; }
}

__global__ void fill_f32_kernel(float* __restrict__ p, float v, int n) {
    int i = blockIdx.x * blockDim.x + threadIdx.x;
    if (i < n) p[i] = v;
}
__global__ void fill_u32_kernel(unsigned* __restrict__ p, unsigned v, int n) {
    int i = blockIdx.x * blockDim.x + threadIdx.x;
    if (i < n) p[i] = v;
}

// Pass 1: per-edge LeakyReLU score -> segment max (ordered-uint atomicMax).
__global__ void edge_max_kernel(const int* __restrict__ ei,
                                const float* __restrict__ s_src,
                                const float* __restrict__ s_dst,
                                unsigned* __restrict__ mord)
{
    int e = blockIdx.x * blockDim.x + threadIdx.x;
    if (e >= ET) return;
    int s, d; edge_nodes(e, ei, s, d);
    float v = s_src[s] + s_dst[d];
    v = (v > 0.f) ? v : NEG_SLOPE * v;
    atomicMax(&mord[d], f2ord(v));
}

// Pass 2: ex = exp(e - max[dst]); denom[dst] += ex.
__global__ void edge_exp_kernel(const int* __restrict__ ei,
                                const float* __restrict__ s_src,
                                const float* __restrict__ s_dst,
                                const unsigned* __restrict__ mord,
                                float* __restrict__ denom,
                                float* __restrict__ ex)
{
    int e = blockIdx.x * blockDim.x + threadIdx.x;
    if (e >= ET) return;
    int s, d; edge_nodes(e, ei, s, d);
    float v = s_src[s] + s_dst[d];
    v = (v > 0.f) ? v : NEG_SLOPE * v;
    float exv = __expf(v - ord2f(mord[d]));
    ex[e] = exv;
    atomicAdd(&denom[d], exv);
}

// Pass 3: agg[dst] += (ex/denom[dst]) * h[src].
// One wave per edge; lane handles feats {2*lane, 2*lane+1} -> one b64 gather.
__global__ __launch_bounds__(256) void edge_agg_kernel(
    const int* __restrict__ ei, const float* __restrict__ h,
    const float* __restrict__ ex, const float* __restrict__ denom,
    float* __restrict__ agg)
{
    const int wave = (blockIdx.x * blockDim.x + threadIdx.x) >> 5;
    const int lane = threadIdx.x & 31;
    if (wave >= ET) return;
    int s, d; edge_nodes(wave, ei, s, d);
    const float alpha = ex[wave] / denom[d];
    v2f hv = *(const v2f*)(h + (size_t)s * 64 + 2 * lane);
    float* dstp = agg + (size_t)d * 64 + 2 * lane;
    atomicAdd(dstp,     alpha * hv.x);
    atomicAdd(dstp + 1, alpha * hv.y);
}

// ---------------------------------------------------------------------------
extern "C" void kernel_launch(void* const* d_in, const int* in_sizes, int n_in,
                              void* d_out, int out_size, void* d_ws, size_t ws_size,
                              hipStream_t stream)
{
    const float* x        = (const float*)d_in[0];
    const int*   ei       = (const int*)  d_in[1];
    const float* gat0_W   = (const float*)d_in[2];
    const float* gat0_as  = (const float*)d_in[3];
    const float* gat0_ad  = (const float*)d_in[4];
    const float* gat0_b   = (const float*)d_in[5];
    const float* gat1_W   = (const float*)d_in[6];
    const float* gat1_as  = (const float*)d_in[7];
    const float* gat1_ad  = (const float*)d_in[8];
    const float* gat1_b   = (const float*)d_in[9];
    const float* lin0_W   = (const float*)d_in[10];
    const float* lin0_b   = (const float*)d_in[11];
    const float* lin1_W   = (const float*)d_in[12];
    const float* lin1_b   = (const float*)d_in[13];
    float* out = (float*)d_out;

    float*    ws    = (float*)d_ws;
    float*    h     = ws + OFF_H;
    float*    aggA  = ws + OFF_AGGA;
    float*    aggB  = ws + OFF_AGGB;
    float*    ssrc  = ws + OFF_SS;
    float*    sdst  = ws + OFF_SD;
    unsigned* mord  = (unsigned*)(ws + OFF_M);
    float*    den   = ws + OFF_DEN;
    float*    ex    = ws + OFF_EX;

    const int strips = NN / 16;                  // 3125 (exact)
    const dim3 gemmGrid((strips + 7) / 8);       // 8 waves / 256-thread block
    const dim3 nodeWaves((NN + 7) / 8);
    const dim3 edgeThreads((ET + 255) / 256);
    const dim3 edgeWaves((ET + 7) / 8);
    const dim3 fillN((NN + 255) / 256);
    const dim3 fillNF((NN * 64 + 255) / 256);

    // ---- shared tail of one GAT layer (after h is computed) ----
    auto gat_softmax_agg = [&](const float* att_s, const float* att_d, float* agg) {
        scores_kernel<<<nodeWaves, 256, 0, stream>>>(h, att_s, att_d, ssrc, sdst, NN);
        fill_u32_kernel<<<fillN, 256, 0, stream>>>(mord, ORD_NEG_INF, NN);
        fill_f32_kernel<<<fillN, 256, 0, stream>>>(den, 0.f, NN);
        fill_f32_kernel<<<fillNF, 256, 0, stream>>>(agg, 0.f, NN * 64);
        edge_max_kernel<<<edgeThreads, 256, 0, stream>>>(ei, ssrc, sdst, mord);
        edge_exp_kernel<<<edgeThreads, 256, 0, stream>>>(ei, ssrc, sdst, mord, den, ex);
        edge_agg_kernel<<<edgeWaves, 256, 0, stream>>>(ei, h, ex, den, agg);
    };

    // GAT layer 0: h = x @ W0 ; aggregate -> aggA (gat0_b folded into next GEMM)
    gemm_wmma_kernel<4, F_IN, HIDF, false, false><<<gemmGrid, 256, 0, stream>>>(
        x, gat0_W, nullptr, nullptr, h, NN);
    gat_softmax_agg(gat0_as, gat0_ad, aggA);

    // GAT layer 1: h = (aggA + gat0_b) @ W1 ; aggregate -> aggB
    gemm_wmma_kernel<4, HIDF, HIDF, true, false><<<gemmGrid, 256, 0, stream>>>(
        aggA, gat1_W, gat0_b, nullptr, h, NN);
    gat_softmax_agg(gat1_as, gat1_ad, aggB);

    // lin0: relu((aggB + gat1_b) @ lin0_W + lin0_b) -> h (reused)
    gemm_wmma_kernel<4, HIDF, HIDF, true, true><<<gemmGrid, 256, 0, stream>>>(
        aggB, lin0_W, gat1_b, lin0_b, h, NN);
    // lin1: relu(h @ lin1_W + lin1_b) -> out (50000 x 10, N-tile zero-padded)
    gemm_wmma_kernel<1, HIDF, NCLS, false, true><<<gemmGrid, 256, 0, stream>>>(
        h, lin1_W, nullptr, lin1_b, out, NN);

    (void)in_sizes; (void)n_in; (void)out_size; (void)ws_size;
}